// AttentionLayer_79869211836828
// MI455X (gfx1250) — compile-verified
//
#include <hip/hip_runtime.h>

// Flash-attention style causal attention with threshold, fp32 WMMA (CDNA5 / gfx1250).
// B=16, S=2048, D=128.

#define B_ 16
#define S_ 2048
#define D_ 128
#define THR 0.3f

typedef __attribute__((ext_vector_type(2))) float v2f;
typedef __attribute__((ext_vector_type(8))) float v8f;

#define KSTRIDE 132   // 16 rows x 128 d, padded: conflict-free b64 reads
#define VSTRIDE 20    // 128 rows (d) x 16 keys, padded
#define PSTRIDE 20    // per-wave 16x16 P scratch, padded

static __device__ __forceinline__
v8f wmma_f32(v2f a, v2f b, v8f c) {
    return __builtin_amdgcn_wmma_f32_16x16x4_f32(false, a, false, b,
                                                 (short)0, c, false, false);
}

__global__ __launch_bounds__(128)
void attn_fa_kernel(const float* __restrict__ Q,
                    const float* __restrict__ T,
                    const float* __restrict__ V,
                    float* __restrict__ O)
{
    __shared__ float lds_k [16  * KSTRIDE];   // K tile, row-major [key][d]
    __shared__ float lds_vt[128 * VSTRIDE];   // V tile, transposed [d][key]
    __shared__ float lds_p [4 * 16 * PSTRIDE];// per-wave P scratch [row][col]

    const int b    = blockIdx.y;
    const int g    = blockIdx.x;       // query group of 64
    const int tid  = threadIdx.x;
    const int wave = tid >> 5;
    const int lane = tid & 31;
    const int n    = lane & 15;        // column / key-in-tile index
    const int hi   = lane >> 4;        // half-wave select

    const int q0     = g * 64 + wave * 16;  // this wave's first query row
    const int myDiag = q0 >> 4;             // last key tile this wave touches
    const int jmax   = g * 4 + 3;           // last key tile any wave in WG touches

    // 1/sqrt(128) * log2(e): softmax via exact v_exp_f32 (2^x)
    const float SCALE2 = 0.08838834764831845f * 1.4426950408889634f;

    // ---- Preload Q tile into WMMA-A layout registers ----
    // A 16x4 f32 layout: lane m=lane&15 holds {Q[m][k0+2*hi], Q[m][k0+2*hi+1]}
    v2f qa[32];
    {
        const float* qp = Q + ((size_t)b * S_ + q0 + n) * D_ + 2 * hi;
#pragma unroll
        for (int c = 0; c < 32; ++c)
            qa[c] = *(const v2f*)(qp + 4 * c);
    }

    v8f y[8];
#pragma unroll
    for (int nt = 0; nt < 8; ++nt) y[nt] = {};
    float m_run[8], l_run[8];
#pragma unroll
    for (int e = 0; e < 8; ++e) { m_run[e] = -1e30f; l_run[e] = 0.0f; }

    for (int j = 0; j <= jmax; ++j) {
        // ---- Cooperative tile load: K row-major, V transposed ----
        {
            const int r     = tid >> 3;        // 0..15 key row
            const int dbase = (tid & 7) * 16;  // 0..112
            const float* kp = T + ((size_t)b * S_ + (size_t)j * 16 + r) * D_ + dbase;
            const float* vp = V + ((size_t)b * S_ + (size_t)j * 16 + r) * D_ + dbase;
#pragma unroll
            for (int i = 0; i < 4; ++i) {
                float4 kq = *(const float4*)(kp + 4 * i);
                *(float4*)&lds_k[r * KSTRIDE + dbase + 4 * i] = kq;
                float4 vq = *(const float4*)(vp + 4 * i);
                lds_vt[(dbase + 4*i + 0) * VSTRIDE + r] = vq.x;
                lds_vt[(dbase + 4*i + 1) * VSTRIDE + r] = vq.y;
                lds_vt[(dbase + 4*i + 2) * VSTRIDE + r] = vq.z;
                lds_vt[(dbase + 4*i + 3) * VSTRIDE + r] = vq.w;
            }
        }
        __syncthreads();

        if (j <= myDiag) {
            // ---- Scores S = Q * K^T (fp32 WMMA, software-pipelined B fetch) ----
            // B 4x16 layout: b.x = Kt[k0+2*hi][n] = T[n][k0+2*hi] -> contiguous f2
            const float* krow = &lds_k[n * KSTRIDE + 2 * hi];
            v8f s0 = {}, s1 = {};
            v2f kb[2][8];
#pragma unroll
            for (int i = 0; i < 8; ++i)
                kb[0][i] = *(const v2f*)(krow + 4 * i);
#pragma unroll
            for (int grp = 0; grp < 4; ++grp) {
                if (grp < 3) {
#pragma unroll
                    for (int i = 0; i < 8; ++i)
                        kb[(grp + 1) & 1][i] =
                            *(const v2f*)(krow + 4 * (8 * (grp + 1) + i));
                }
#pragma unroll
                for (int i = 0; i < 8; i += 2) {
                    s0 = wmma_f32(qa[8 * grp + i],     kb[grp & 1][i],     s0);
                    s1 = wmma_f32(qa[8 * grp + i + 1], kb[grp & 1][i + 1], s1);
                }
            }

            // ---- Threshold, causal mask, scale, online softmax ----
            const int kk = j * 16 + n;   // global key index of this lane's column
            float al[8], p[8];
#pragma unroll
            for (int e = 0; e < 8; ++e) {
                float sv = s0[e] + s1[e];
                sv = (sv > THR) ? sv : 0.0f;          // tf.where threshold (false -> 0)
                float z = sv * SCALE2;
                const int m = e + 8 * hi;             // row within 16-row tile
                if (kk > q0 + m) z = -1e30f;          // strict causal mask
                // row max across the 16 lanes holding this row
                float rm = z;
                rm = fmaxf(rm, __shfl_xor(rm, 1, 32));
                rm = fmaxf(rm, __shfl_xor(rm, 2, 32));
                rm = fmaxf(rm, __shfl_xor(rm, 4, 32));
                rm = fmaxf(rm, __shfl_xor(rm, 8, 32));
                float mn = fmaxf(m_run[e], rm);
                float a  = __builtin_amdgcn_exp2f(m_run[e] - mn);
                float pe = __builtin_amdgcn_exp2f(z - mn);
                float rs = pe;
                rs += __shfl_xor(rs, 1, 32);
                rs += __shfl_xor(rs, 2, 32);
                rs += __shfl_xor(rs, 4, 32);
                rs += __shfl_xor(rs, 8, 32);
                l_run[e] = l_run[e] * a + rs;
                m_run[e] = mn;
                al[e] = a;
                p[e]  = pe;
            }

            // rescale running output by alpha (per-row)
#pragma unroll
            for (int nt = 0; nt < 8; ++nt)
#pragma unroll
                for (int e = 0; e < 8; ++e)
                    y[nt][e] *= al[e];

            // ---- C-layout -> A-layout for P via per-wave LDS scratch ----
            float* pp = &lds_p[wave * 16 * PSTRIDE];
#pragma unroll
            for (int e = 0; e < 8; ++e)
                pp[(e + 8 * hi) * PSTRIDE + n] = p[e];  // P[row][col]

            v2f pa[4];
#pragma unroll
            for (int c = 0; c < 4; ++c)
                pa[c] = *(const v2f*)&pp[n * PSTRIDE + 4 * c + 2 * hi];

            // ---- y += P * V : 4 independent chains per half, batched B fetch ----
#pragma unroll
            for (int half = 0; half < 2; ++half) {
                v2f vb[16];
#pragma unroll
                for (int nt = 0; nt < 4; ++nt)
#pragma unroll
                    for (int c = 0; c < 4; ++c)
                        vb[nt * 4 + c] = *(const v2f*)
                            &lds_vt[((half * 4 + nt) * 16 + n) * VSTRIDE + 4 * c + 2 * hi];
#pragma unroll
                for (int nt = 0; nt < 4; ++nt) {
                    v8f acc = y[half * 4 + nt];
#pragma unroll
                    for (int c = 0; c < 4; ++c)
                        acc = wmma_f32(pa[c], vb[nt * 4 + c], acc);
                    y[half * 4 + nt] = acc;
                }
            }
        }
        __syncthreads();
    }

    // ---- Normalize and store ----
    float inv[8];
#pragma unroll
    for (int e = 0; e < 8; ++e) inv[e] = 1.0f / l_run[e];
    float* op = O + ((size_t)b * S_ + q0) * D_;
#pragma unroll
    for (int nt = 0; nt < 8; ++nt)
#pragma unroll
        for (int e = 0; e < 8; ++e)
            op[(e + 8 * hi) * D_ + nt * 16 + n] = y[nt][e] * inv[e];
}

extern "C" void kernel_launch(void* const* d_in, const int* in_sizes, int n_in,
                              void* d_out, int out_size, void* d_ws, size_t ws_size,
                              hipStream_t stream) {
    const float* Q = (const float*)d_in[0];
    const float* T = (const float*)d_in[1];
    const float* V = (const float*)d_in[2];
    float*       O = (float*)d_out;
    dim3 grid(S_ / 64, B_);   // 32 query groups x 16 batches
    attn_fa_kernel<<<grid, dim3(128), 0, stream>>>(Q, T, V, O);
}